// KnnAttentionLayer_54254026883702
// MI455X (gfx1250) — compile-verified
//
#include <hip/hip_runtime.h>
#include <math.h>

#define B_   2
#define N_   8192
#define K_   16
#define H_   4
#define CIN  64
#define COUT 128
#define HD_  128
#define PTS  8      // points per block (1 per wave)

typedef float v2f __attribute__((ext_vector_type(2)));
typedef float v8f __attribute__((ext_vector_type(8)));

__device__ __forceinline__ v8f wmma4(v2f a, v2f b, v8f c) {
  // V_WMMA_F32_16X16X4_F32 : D = A(16x4) * B(4x16) + C(16x16), fp32
  return __builtin_amdgcn_wmma_f32_16x16x4_f32(
      /*neg_a=*/false, a, /*neg_b=*/false, b,
      /*c_mod=*/(short)0, c, /*reuse_a=*/false, /*reuse_b=*/false);
}

__global__ __launch_bounds__(256) void knn_attn_kernel(
    const float* __restrict__ xyzs, const float* __restrict__ features,
    const int*   __restrict__ kg,
    const float* __restrict__ Wk,  const float* __restrict__ Wv,
    const float* __restrict__ Wq,
    const float* __restrict__ A1,  const float* __restrict__ b1,
    const float* __restrict__ A2,  const float* __restrict__ b2,
    const float* __restrict__ P1,  const float* __restrict__ bp1,
    const float* __restrict__ P2,  const float* __restrict__ bp2,
    const float* __restrict__ Wout,const float* __restrict__ bout,
    float* __restrict__ out)
{
  __shared__ __align__(16) float qf[16 * CIN];            // 4 KB  (rows 8..15 zero)
  __shared__ float qxyz[PTS * 3];
  __shared__ int   idxbuf[PTS * K_];
  __shared__ __align__(16) float hbuf[PTS * 16 * 4];      // 2 KB  pos-MLP hidden, K padded to 4
  __shared__ __align__(16) float nbrF[PTS * 16 * CIN];    // 32 KB gathered neighbor features
  __shared__ __align__(16) float qmat[PTS * HD_];         // 4 KB  queries
  __shared__ __align__(16) float S[PTS * 16 * HD_];       // 64 KB attn_in then attn_mid
  __shared__ float wbuf[PTS * 16 * H_];                   // 2 KB  softmax weights
  __shared__ __align__(16) float fusedBuf[16 * HD_];      // 8 KB  (rows 8..15 zero)
  __shared__ __align__(16) float A2t[16 * HD_];           // 8 KB  A2 transposed, cols 4..15 zero
  __shared__ __align__(16) float P2t[HD_ * 4];            // 2 KB  P2 transposed, K padded to 4
  __shared__ float b2s[16];                               // b2 zero-padded

  const int tid  = threadIdx.x;
  const int wv   = tid >> 5;      // wave id = point id within block
  const int lane = tid & 31;
  const int nblk = (int)blockIdx.x;               // 0 .. B*N/PTS-1
  const int bb   = nblk / (N_ / PTS);
  const int pbase = (nblk % (N_ / PTS)) * PTS;

  const int m   = lane & 15;        // A row
  const int n   = lane & 15;        // B col
  const int kb2 = (lane >> 4) * 2;  // K sub-offset
  const int rhi = 8 * (lane >> 4);  // C/D row offset

  // ---- phase 0: cooperative loads + zero padding + weight staging -----
  {
    const float* src = features + ((size_t)bb * N_ + pbase) * CIN;
    for (int i = tid; i < PTS * CIN; i += 256) { qf[i] = src[i]; qf[PTS * CIN + i] = 0.f; }
    const float* sx = xyzs + ((size_t)bb * N_ + pbase) * 3;
    if (tid < PTS * 3) qxyz[tid] = sx[tid];
    const int* sg = kg + ((size_t)bb * N_ + pbase) * K_;
    if (tid < PTS * K_) idxbuf[tid] = sg[tid];
    // zero pad rows of fusedBuf (rows 8..15 never written later)
    for (int i = tid; i < 8 * HD_; i += 256) fusedBuf[8 * HD_ + i] = 0.f;
    // A2 transposed + zero pad: A2t[n][k] = (n<4) ? A2[k][n] : 0
    for (int i = tid; i < 16 * HD_; i += 256) {
      int r = i >> 7, k = i & (HD_ - 1);
      A2t[i] = (r < H_) ? A2[k * H_ + r] : 0.f;
    }
    // P2 transposed + zero pad: P2t[c][j] = (j<3) ? P2[j][c] : 0
    if (tid < HD_) {
      P2t[tid * 4 + 0] = P2[0 * HD_ + tid];
      P2t[tid * 4 + 1] = P2[1 * HD_ + tid];
      P2t[tid * 4 + 2] = P2[2 * HD_ + tid];
      P2t[tid * 4 + 3] = 0.f;
    }
    if (tid < 16) b2s[tid] = (tid < H_) ? b2[tid] : 0.f;
  }
  __syncthreads();

  float* nf = nbrF + wv * 16 * CIN;
  float* hb = hbuf + wv * 16 * 4;
  float* sw = S    + wv * 16 * HD_;

  // ---- phase 1: per-wave gather + tiny pos MLP + query tile -----------
  {
    // gather 16 x 64 neighbor features (2 rows / iter, float4 lanes)
    for (int it = 0; it < 8; ++it) {
      int r = it * 2 + (lane >> 4);
      int g = idxbuf[wv * K_ + r];
      const float4 v4 = *(const float4*)(features + ((size_t)bb * N_ + g) * CIN + (lane & 15) * 4);
      *(float4*)(nf + r * CIN + (lane & 15) * 4) = v4;
    }
    // neighbor xyz, rel_pos, h = relu(rel@P1+bp1) padded to K=4
    if (lane < 16) {
      int g = idxbuf[wv * K_ + lane];
      const float* nx = xyzs + ((size_t)bb * N_ + g) * 3;
      float rx = qxyz[wv * 3 + 0] - nx[0];
      float ry = qxyz[wv * 3 + 1] - nx[1];
      float rz = qxyz[wv * 3 + 2] - nx[2];
      float h0 = fmaxf(rx * P1[0] + ry * P1[3] + rz * P1[6] + bp1[0], 0.f);
      float h1 = fmaxf(rx * P1[1] + ry * P1[4] + rz * P1[7] + bp1[1], 0.f);
      float h2 = fmaxf(rx * P1[2] + ry * P1[5] + rz * P1[8] + bp1[2], 0.f);
      hb[lane * 4 + 0] = h0; hb[lane * 4 + 1] = h1;
      hb[lane * 4 + 2] = h2; hb[lane * 4 + 3] = 0.f;
    }
    // queries: qmat[:, wv*16 .. +16] = qf[16(8 valid),64] @ Wq
    v2f aq[16];
    #pragma unroll
    for (int ks = 0; ks < 16; ++ks)
      aq[ks] = *(const v2f*)(qf + m * CIN + ks * 4 + kb2);
    v8f acc = {};
    #pragma unroll
    for (int ks = 0; ks < 16; ++ks) {
      v2f bf;
      bf.x = Wq[(ks * 4 + kb2)     * HD_ + wv * 16 + n];
      bf.y = Wq[(ks * 4 + kb2 + 1) * HD_ + wv * 16 + n];
      acc = wmma4(aq[ks], bf, acc);
    }
    if (lane < 16) {
      #pragma unroll
      for (int v = 0; v < 8; ++v) qmat[v * HD_ + wv * 16 + lane] = acc[v];
    }
  }
  __syncthreads();

  // ---- phase 2: keys + attn_in = q - keys + pe ------------------------
  {
    v2f af[16];
    #pragma unroll
    for (int ks = 0; ks < 16; ++ks)
      af[ks] = *(const v2f*)(nf + m * CIN + ks * 4 + kb2);
    v2f ah = *(const v2f*)(hb + m * 4 + kb2);
    for (int nt = 0; nt < 8; ++nt) {
      int c = nt * 16 + n;
      v2f bh = *(const v2f*)(P2t + c * 4 + kb2);
      v8f pe = {};
      pe = wmma4(ah, bh, pe);
      v8f kacc = {};
      #pragma unroll
      for (int ks = 0; ks < 16; ++ks) {
        v2f bf;
        bf.x = Wk[(ks * 4 + kb2)     * HD_ + c];
        bf.y = Wk[(ks * 4 + kb2 + 1) * HD_ + c];
        kacc = wmma4(af[ks], bf, kacc);
      }
      float qv   = qmat[wv * HD_ + c];
      float bias = bp2[c];
      #pragma unroll
      for (int v = 0; v < 8; ++v)
        sw[(v + rhi) * HD_ + c] = qv - kacc[v] + pe[v] + bias;
    }
  }

  // ---- phase 3: attn_mid = relu(attn_in @ A1 + b1), in-place in slab --
  {
    v2f af2[32];                      // full attn_in A-fragments captured
    #pragma unroll
    for (int ks = 0; ks < 32; ++ks)
      af2[ks] = *(const v2f*)(sw + m * HD_ + ks * 4 + kb2);
    for (int nt = 0; nt < 8; ++nt) {
      int c = nt * 16 + n;
      v8f acc = {};
      #pragma unroll
      for (int ks = 0; ks < 32; ++ks) {
        v2f bf;
        bf.x = A1[(ks * 4 + kb2)     * HD_ + c];
        bf.y = A1[(ks * 4 + kb2 + 1) * HD_ + c];
        acc = wmma4(af2[ks], bf, acc);
      }
      float bv = b1[c];
      #pragma unroll
      for (int v = 0; v < 8; ++v)
        sw[(v + rhi) * HD_ + c] = fmaxf(acc[v] + bv, 0.f);   // safe: attn_in lives in af2
    }
  }

  // ---- phase 4: logits = attn_mid @ A2t, softmax over 16 neighbors ----
  {
    v2f af3[32];
    #pragma unroll
    for (int ks = 0; ks < 32; ++ks)
      af3[ks] = *(const v2f*)(sw + m * HD_ + ks * 4 + kb2);
    v8f lac = {};
    #pragma unroll
    for (int ks = 0; ks < 32; ++ks) {
      v2f bf = *(const v2f*)(A2t + n * HD_ + ks * 4 + kb2);  // padded, unconditional
      lac = wmma4(af3[ks], bf, lac);
    }
    float bv = b2s[n];
    float lv[8];
    float mx = -1e30f;
    #pragma unroll
    for (int v = 0; v < 8; ++v) { lv[v] = lac[v] + bv; mx = fmaxf(mx, lv[v]); }
    mx = fmaxf(mx, __shfl_xor(mx, 16, 32));       // 16 rows split across lane pairs
    float s = 0.f;
    #pragma unroll
    for (int v = 0; v < 8; ++v) { lv[v] = __expf(lv[v] - mx); s += lv[v]; }
    s += __shfl_xor(s, 16, 32);
    float inv = 1.f / s;
    if (n < H_) {
      #pragma unroll
      for (int v = 0; v < 8; ++v) wbuf[wv * 64 + (v + rhi) * H_ + n] = lv[v] * inv;
    }
  }

  // ---- phase 5: recompute values+pe, weighted reduce over neighbors ---
  {
    v2f af[16];
    #pragma unroll
    for (int ks = 0; ks < 16; ++ks)
      af[ks] = *(const v2f*)(nf + m * CIN + ks * 4 + kb2);
    v2f ah = *(const v2f*)(hb + m * 4 + kb2);
    for (int nt = 0; nt < 8; ++nt) {
      int c = nt * 16 + n;
      v2f bh = *(const v2f*)(P2t + c * 4 + kb2);
      v8f vac = {};
      vac = wmma4(ah, bh, vac);                    // pe
      #pragma unroll
      for (int ks = 0; ks < 16; ++ks) {
        v2f bf;
        bf.x = Wv[(ks * 4 + kb2)     * HD_ + c];
        bf.y = Wv[(ks * 4 + kb2 + 1) * HD_ + c];
        vac = wmma4(af[ks], bf, vac);              // + values
      }
      float bias = bp2[c];
      int   h    = nt >> 1;                        // head = c >> 5
      float part = 0.f;
      #pragma unroll
      for (int v = 0; v < 8; ++v) {
        float wgt = wbuf[wv * 64 + (v + rhi) * H_ + h];
        part += wgt * (vac[v] + bias);
      }
      part += __shfl_xor(part, 16, 32);
      if (lane < 16) fusedBuf[wv * HD_ + c] = part;
    }
  }
  __syncthreads();

  // ---- phase 6: out = fused[16(8 valid),128] @ Wout + bout ------------
  {
    v2f af4[32];
    #pragma unroll
    for (int ks = 0; ks < 32; ++ks)
      af4[ks] = *(const v2f*)(fusedBuf + m * HD_ + ks * 4 + kb2);   // padded rows -> 0
    v8f acc = {};
    #pragma unroll
    for (int ks = 0; ks < 32; ++ks) {
      v2f bf;
      bf.x = Wout[(ks * 4 + kb2)     * COUT + wv * 16 + n];
      bf.y = Wout[(ks * 4 + kb2 + 1) * COUT + wv * 16 + n];
      acc = wmma4(af4[ks], bf, acc);
    }
    if (lane < 16) {
      float bo = bout[wv * 16 + lane];
      #pragma unroll
      for (int v = 0; v < 8; ++v)
        out[((size_t)bb * N_ + pbase + v) * COUT + wv * 16 + lane] = acc[v] + bo;
    }
  }
}

extern "C" void kernel_launch(void* const* d_in, const int* in_sizes, int n_in,
                              void* d_out, int out_size, void* d_ws, size_t ws_size,
                              hipStream_t stream) {
  (void)in_sizes; (void)n_in; (void)d_ws; (void)ws_size; (void)out_size;
  const float* xyzs     = (const float*)d_in[0];
  const float* features = (const float*)d_in[1];
  const int*   kg       = (const int*)  d_in[2];
  const float* Wk       = (const float*)d_in[3];
  const float* Wv       = (const float*)d_in[4];
  const float* Wq       = (const float*)d_in[5];
  const float* A1       = (const float*)d_in[6];
  const float* b1       = (const float*)d_in[7];
  const float* A2       = (const float*)d_in[8];
  const float* b2       = (const float*)d_in[9];
  const float* P1       = (const float*)d_in[10];
  const float* bp1      = (const float*)d_in[11];
  const float* P2       = (const float*)d_in[12];
  const float* bp2      = (const float*)d_in[13];
  const float* Wout     = (const float*)d_in[14];
  const float* bout     = (const float*)d_in[15];
  float* out = (float*)d_out;

  dim3 grid(B_ * N_ / PTS);
  dim3 block(256);
  hipLaunchKernelGGL(knn_attn_kernel, grid, block, 0, stream,
                     xyzs, features, kg, Wk, Wv, Wq, A1, b1, A2, b2,
                     P1, bp1, P2, bp2, Wout, bout, out);
}